// SelectedAttention_1219770712405
// MI455X (gfx1250) — compile-verified
//
#include <hip/hip_runtime.h>
#include <hip/hip_bf16.h>

#define S_LEN  4096
#define D_DIM  128
#define TOPK   64
#define QTILE  16
#define NWAVES 8
#define WAVESZ 32
#define MAXL   96   // compact-list capacity per wave (>=64 with tie slack)

typedef __attribute__((ext_vector_type(2))) float v2f;
typedef __attribute__((ext_vector_type(8))) float v8f;

// order-preserving float <-> uint key (total order, -0 < +0)
__device__ __forceinline__ unsigned f2key(float f) {
  unsigned u = __float_as_uint(f);
  return (u & 0x80000000u) ? ~u : (u | 0x80000000u);
}
__device__ __forceinline__ float key2f(unsigned k) {
  unsigned u = (k & 0x80000000u) ? (k ^ 0x80000000u) : ~k;
  return __uint_as_float(u);
}

// per-batch column sums of V: vsum[b][d] = sum_k V[b][k][d]
__global__ void vsum_kernel(const float* __restrict__ V, float* __restrict__ vsum) {
  int b = blockIdx.x, d = threadIdx.x;
  const float* vp = V + (size_t)b * S_LEN * D_DIM + d;
  float s = 0.f;
  for (int k = 0; k < S_LEN; ++k) s += vp[(size_t)k * D_DIM];
  vsum[b * D_DIM + d] = s;
}

__global__ __launch_bounds__(NWAVES * WAVESZ)
void selattn_kernel(const float* __restrict__ Q, const float* __restrict__ K,
                    const float* __restrict__ V, const float* __restrict__ vsum,
                    float* __restrict__ out) {
  extern __shared__ char smem[];
  unsigned* keys  = (unsigned*)smem;                          // [QTILE][S_LEN]
  unsigned* listK = (unsigned*)(smem + QTILE * S_LEN * 4);    // [NWAVES][MAXL]
  float*    listE = (float*)(listK + NWAVES * MAXL);          // [NWAVES][MAXL]
  int*      cnts  = (int*)(listE + NWAVES * MAXL);            // [NWAVES]

  const int b      = blockIdx.x / (S_LEN / QTILE);
  const int q_base = (blockIdx.x % (S_LEN / QTILE)) * QTILE;
  const int lane   = threadIdx.x & (WAVESZ - 1);
  const int wav    = threadIdx.x / WAVESZ;

  // ---------------- pass 1: scores tile via fp32 WMMA ----------------
  // A (16x4 f32) frag: lane<16 -> M=lane, K={0,1}; lane>=16 -> M=lane-16, K={2,3}
  const int m  = lane & 15;
  const int kp = (lane >> 4) * 2;
  const float* qrow = Q + (size_t)(b * S_LEN + q_base + m) * D_DIM + kp;
  v2f aF[32];
#pragma unroll
  for (int j = 0; j < 32; ++j) aF[j] = *(const v2f*)(qrow + 4 * j);

  const int row0 = (lane >> 4) * 8;
  for (int kt = 0; kt < 32; kt += 2) {            // 2 key tiles in flight per wave
    const int k_base = wav * 512 + kt * 16;
    const float* krow0 = K + (size_t)(b * S_LEN + k_base + m) * D_DIM + kp;
    const float* krow1 = krow0 + (size_t)16 * D_DIM;
    v8f c0 = {}, c1 = {};
#pragma unroll
    for (int j = 0; j < 32; ++j) {                // K-loop over D in steps of 4
      v2f b0 = *(const v2f*)(krow0 + 4 * j);      // B(4x16): lane<16 K={0,1}, else {2,3}
      v2f b1 = *(const v2f*)(krow1 + 4 * j);
      c0 = __builtin_amdgcn_wmma_f32_16x16x4_f32(false, aF[j], false, b0,
                                                 (short)0, c0, false, false);
      c1 = __builtin_amdgcn_wmma_f32_16x16x4_f32(false, aF[j], false, b1,
                                                 (short)0, c1, false, false);
    }
    // C layout: VGPR r, lanes<16 -> M=r, lanes>=16 -> M=r+8; N = lane%16
    const int col0 = k_base + (lane & 15);
#pragma unroll
    for (int r = 0; r < 8; ++r) {
      keys[(row0 + r) * S_LEN + col0]      = f2key(c0[r] * 0.08838834764831845f);
      keys[(row0 + r) * S_LEN + col0 + 16] = f2key(c1[r] * 0.08838834764831845f);
    }
  }
  __syncthreads();

  // ---------------- pass 2+3: per-row top-64 threshold + output -------
  for (int r2 = 0; r2 < QTILE / NWAVES; ++r2) {   // uniform: every wave does 2 rows
    const int row = wav * (QTILE / NWAVES) + r2;
    const unsigned* krow = keys + row * S_LEN;

    // radix binary search: max t with count(key >= t) >= 64  ==  64th largest key
    unsigned t = 0u;
    for (int bit = 31; bit >= 0; --bit) {
      unsigned cand = t | (1u << bit);
      int cnt = 0;
      for (int i = 0; i < S_LEN / (WAVESZ * 4); ++i) {
        uint4 kk = ((const uint4*)krow)[i * WAVESZ + lane];
        cnt += (kk.x >= cand) + (kk.y >= cand) + (kk.z >= cand) + (kk.w >= cand);
      }
#pragma unroll
      for (int off = 16; off; off >>= 1) cnt += __shfl_xor(cnt, off, WAVESZ);
      if (cnt >= TOPK) t = cand;
    }

    if (lane == 0) cnts[wav] = 0;
    __syncthreads();

    // compact selected (k, exp(s)) pairs; accumulate Z locally
    float zloc = 0.f;
    for (int i = 0; i < S_LEN / (WAVESZ * 4); ++i) {
      int base_k = (i * WAVESZ + lane) * 4;
      uint4 kq = ((const uint4*)krow)[i * WAVESZ + lane];
      unsigned kv[4] = {kq.x, kq.y, kq.z, kq.w};
#pragma unroll
      for (int c4 = 0; c4 < 4; ++c4) {
        if (kv[c4] >= t) {
          float e = __expf(key2f(kv[c4]));
          zloc += e;
          int pos = atomicAdd(&cnts[wav], 1);
          if (pos < MAXL) {
            listK[wav * MAXL + pos] = base_k + c4;
            listE[wav * MAXL + pos] = e;
          }
        }
      }
    }
#pragma unroll
    for (int off = 16; off; off >>= 1) zloc += __shfl_xor(zloc, off, WAVESZ);
    __syncthreads();

    // out = (sum (e-1)*V[k] + sum_all V) / (sum e + (S - ns)); 4 dims per lane
    int ns = cnts[wav];
    int n  = ns < MAXL ? ns : MAXL;
    float4 acc = make_float4(0.f, 0.f, 0.f, 0.f);
    for (int i = 0; i < n; ++i) {
      int   ksel = listK[wav * MAXL + i];
      float e1   = listE[wav * MAXL + i] - 1.0f;
      float4 v4  = *(const float4*)(V + (size_t)(b * S_LEN + ksel) * D_DIM + lane * 4);
      acc.x += e1 * v4.x; acc.y += e1 * v4.y; acc.z += e1 * v4.z; acc.w += e1 * v4.w;
    }
    float  invZ = 1.0f / (zloc + (float)(S_LEN - ns));
    float4 vs   = *(const float4*)(vsum + b * D_DIM + lane * 4);
    float4 o;
    o.x = (acc.x + vs.x) * invZ;
    o.y = (acc.y + vs.y) * invZ;
    o.z = (acc.z + vs.z) * invZ;
    o.w = (acc.w + vs.w) * invZ;
    *(float4*)(out + (size_t)(b * S_LEN + q_base + row) * D_DIM + lane * 4) = o;
    __syncthreads();
  }
}

extern "C" void kernel_launch(void* const* d_in, const int* in_sizes, int n_in,
                              void* d_out, int out_size, void* d_ws, size_t ws_size,
                              hipStream_t stream) {
  const float* Q = (const float*)d_in[0];
  const float* K = (const float*)d_in[1];
  const float* V = (const float*)d_in[2];
  float* out  = (float*)d_out;
  float* vsum = (float*)d_ws;                     // B*D floats of scratch
  const int B = in_sizes[0] / (S_LEN * D_DIM);

  const size_t smem = (size_t)QTILE * S_LEN * 4          // score keys (256 KB)
                    + (size_t)NWAVES * MAXL * 4 * 2      // compact lists
                    + (size_t)NWAVES * 4;                // counters
  (void)hipFuncSetAttribute((const void*)selattn_kernel,
                            hipFuncAttributeMaxDynamicSharedMemorySize, (int)smem);

  vsum_kernel<<<B, D_DIM, 0, stream>>>(V, vsum);
  selattn_kernel<<<B * (S_LEN / QTILE), NWAVES * WAVESZ, smem, stream>>>(Q, K, V, vsum, out);
}